// VLLMKVCache_88356067213998
// MI455X (gfx1250) — compile-verified
//
#include <hip/hip_runtime.h>

// ---------------------------------------------------------------------------
// VLLM KV-cache insert: out = cache; out[slot(t)] = input[t]
// Pure data movement (~1.14 GiB total) -> HBM-bound, floor ~52us @ 23.3 TB/s.
// Kernel 1: bulk cache->out copy via gfx1250 async global<->LDS DMA path
//           (GLOBAL_LOAD_ASYNC_TO_LDS_B128 / GLOBAL_STORE_ASYNC_FROM_LDS_B128,
//            ASYNCcnt waits), falling back to non-temporal B128 copy if the
//           async builtins are unavailable on this toolchain.
// Kernel 2: per-token scatter (contiguous 4KiB per token), NT B128.
// ---------------------------------------------------------------------------

typedef float v4f __attribute__((ext_vector_type(4)));

#define CPY_BLOCK 256
#define CPY_UNROLL 4
#define TILE4 (CPY_BLOCK * CPY_UNROLL) // float4 elements per block-tile (16 KiB)

#if defined(__HIP_DEVICE_COMPILE__) &&                                         \
    __has_builtin(__builtin_amdgcn_global_load_async_to_lds_b128) &&           \
    __has_builtin(__builtin_amdgcn_global_store_async_from_lds_b128)
#define USE_ASYNC_COPY 1
#else
#define USE_ASYNC_COPY 0
#endif

#if USE_ASYNC_COPY
// Builtin signature (from clang diagnostic): arg0 = v4i in AS1 (__device__),
// arg1 = v4i in AS3 (__shared__), then imm offset + imm cpol.
typedef int v4i_t __attribute__((ext_vector_type(4)));
typedef __attribute__((address_space(1))) v4i_t gv4i_t; // global
typedef __attribute__((address_space(3))) v4i_t lv4i_t; // LDS
// Generic->AS1: 64-bit value identical. Generic->AS3: flat LDS pointers carry
// the LDS byte offset in addr[31:0] (ISA flat aperture rules), so truncation
// reproduces what addrspacecast emits. AS3 pointers are 32-bit.
#define AS_GLOBAL(p) ((gv4i_t*)(unsigned long long)(p))
#define AS_LOCAL(p) ((lv4i_t*)(unsigned int)(unsigned long long)(p))

#if __has_builtin(__builtin_amdgcn_s_wait_asynccnt)
#define WAIT_ASYNC0() __builtin_amdgcn_s_wait_asynccnt(0)
#else
#define WAIT_ASYNC0() asm volatile("s_wait_asynccnt 0" ::: "memory")
#endif
#endif // USE_ASYNC_COPY

// -------------------------- bulk copy kernel -------------------------------
__global__ __launch_bounds__(CPY_BLOCK) void kvcache_copy_kernel(
    const v4f* __restrict__ src, v4f* __restrict__ dst,
    unsigned long long n4) {
  const unsigned tid = threadIdx.x;
#if USE_ASYNC_COPY
  __shared__ v4f lbuf[TILE4];
  const unsigned long long ntiles = n4 / TILE4; // full tiles only
  for (unsigned long long tile = blockIdx.x; tile < ntiles; tile += gridDim.x) {
    const unsigned long long base = tile * (unsigned long long)TILE4;
    // Issue CPY_UNROLL async B128 loads global -> LDS (512 B/wave each).
#pragma unroll
    for (int u = 0; u < CPY_UNROLL; ++u) {
      const unsigned idx = tid + u * CPY_BLOCK;
      __builtin_amdgcn_global_load_async_to_lds_b128(
          AS_GLOBAL(src + base + idx), AS_LOCAL(&lbuf[idx]), 0, 0);
    }
    // ASYNCcnt: loads vs stores retire out of order, so counted waits across
    // mixed types are unsafe -> drain fully before touching/reusing LDS.
    WAIT_ASYNC0();
#pragma unroll
    for (int u = 0; u < CPY_UNROLL; ++u) {
      const unsigned idx = tid + u * CPY_BLOCK;
      __builtin_amdgcn_global_store_async_from_lds_b128(
          AS_GLOBAL(dst + base + idx), AS_LOCAL(&lbuf[idx]), 0, 0);
    }
    WAIT_ASYNC0(); // stores drained before next iteration reuses lbuf
  }
  // Tail (n4 % TILE4), handled by block 0 with direct NT copies.
  if (blockIdx.x == 0) {
    for (unsigned long long i = ntiles * TILE4 + tid; i < n4; i += CPY_BLOCK) {
      v4f v = __builtin_nontemporal_load(src + i);
      __builtin_nontemporal_store(v, dst + i);
    }
  }
#else
  // Fallback: grid-stride, 4 independent in-flight B128 NT loads per thread.
  const unsigned long long gsz = (unsigned long long)gridDim.x * CPY_BLOCK;
  unsigned long long i = (unsigned long long)blockIdx.x * CPY_BLOCK + tid;
  for (; i + 3 * gsz < n4; i += 4 * gsz) {
    v4f a = __builtin_nontemporal_load(src + i);
    v4f b = __builtin_nontemporal_load(src + i + gsz);
    v4f c = __builtin_nontemporal_load(src + i + 2 * gsz);
    v4f d = __builtin_nontemporal_load(src + i + 3 * gsz);
    __builtin_nontemporal_store(a, dst + i);
    __builtin_nontemporal_store(b, dst + i + gsz);
    __builtin_nontemporal_store(c, dst + i + 2 * gsz);
    __builtin_nontemporal_store(d, dst + i + 3 * gsz);
  }
  for (; i < n4; i += gsz) {
    v4f v = __builtin_nontemporal_load(src + i);
    __builtin_nontemporal_store(v, dst + i);
  }
#endif
}

// -------------------------- token scatter kernel ---------------------------
// input[T, H*D] rows are layout-identical to cache rows -> each token is one
// contiguous (H*D*4)-byte copy to row (block_index*BLOCK_SIZE + block_offset).
__global__ __launch_bounds__(256) void kvcache_scatter_kernel(
    const v4f* __restrict__ inp, v4f* __restrict__ out,
    const int* __restrict__ bidx, const int* __restrict__ boff,
    int ntok, int tok4, int block_size) {
  const long long tid = (long long)blockIdx.x * blockDim.x + threadIdx.x;
  const long long total = (long long)ntok * tok4;
  if (tid >= total) return;
  const int token = (int)(tid / tok4);
  const int j = (int)(tid - (long long)token * tok4);
  const long long slot = (long long)bidx[token] * block_size + boff[token];
  v4f v = __builtin_nontemporal_load(inp + tid);
  __builtin_nontemporal_store(v, out + slot * tok4 + j);
}

extern "C" void kernel_launch(void* const* d_in, const int* in_sizes, int n_in,
                              void* d_out, int out_size, void* d_ws,
                              size_t ws_size, hipStream_t stream) {
  (void)n_in; (void)d_ws; (void)ws_size; (void)out_size;
  const v4f* inp = (const v4f*)d_in[0];     // [T, H, D] f32
  const v4f* cache = (const v4f*)d_in[1];   // [B, BS, H, D] f32
  const int* bidx = (const int*)d_in[4];    // [T]
  const int* boff = (const int*)d_in[5];    // [T]
  v4f* out = (v4f*)d_out;

  const unsigned long long n4 = (unsigned long long)in_sizes[1] / 4;
  const int ntok = in_sizes[4];
  const int tokstride = in_sizes[0] / ntok; // H*D floats (1024)
  const int tok4 = tokstride / 4;           // float4 per token (256)
  const int block_size = 128;               // BLOCK_SIZE per reference

  // 1) out = cache (bulk streaming copy).
  kvcache_copy_kernel<<<4096, CPY_BLOCK, 0, stream>>>(cache, out, n4);

  // 2) out[slot(t)] = input[t] (stream-ordered after the copy).
  const long long total = (long long)ntok * tok4;
  const int blocks = (int)((total + 255) / 256);
  kvcache_scatter_kernel<<<blocks, 256, 0, stream>>>(inp, out, bidx, boff,
                                                     ntok, tok4, block_size);
}